// AttentionBlock_2482491097642
// MI455X (gfx1250) — compile-verified
//
#include <hip/hip_runtime.h>
#include <hip/hip_bf16.h>

// MI455X / gfx1250, wave32. All heavy math on V_WMMA_F32_16X16X4_F32 (exact fp32).

typedef __attribute__((ext_vector_type(2))) float v2f;
typedef __attribute__((ext_vector_type(8))) float v8f;

#define WMMA_F32(a, b, c) \
  __builtin_amdgcn_wmma_f32_16x16x4_f32(false, (a), false, (b), (short)0, (c), false, false)

static constexpr int Bn = 8;
static constexpr int Sn = 2048;
static constexpr int Dn = 1024;

// ---------------------------------------------------------------------------
// GEMM: Out[m,n] = act( sum_k X[m,k] * W[n,k] + bias[n] ) (+ Res[m,n])
// M = 16384, N = K = 1024.  Block tile 128x128, K-chunk 32, 8 waves,
// each wave a 32x64 register tile (2x4 fragments). Double-buffered LDS:
// loads for chunk i+1 issue before the 64-WMMA compute of chunk i, so the
// loadcnt wait lands after the matrix work instead of serializing the fill.
// ---------------------------------------------------------------------------
__global__ __launch_bounds__(256) void gemm_wmma(
    const float* __restrict__ X, const float* __restrict__ W,
    const float* __restrict__ bias, const float* __restrict__ Res,
    float* __restrict__ Out, int relu_res) {
  __shared__ float sX[2][128 * 32];
  __shared__ float sW[2][128 * 32];

  const int m0 = blockIdx.x * 128;
  const int n0 = blockIdx.y * 128;
  const int t = threadIdx.x;
  const int wave = t >> 5;
  const int lane = t & 31;
  const int lm = lane & 15;   // M row (A) / N col (B) within 16
  const int lg = lane >> 4;   // selects K pair (A/B) and M half (C)
  const int wm = (wave >> 1) * 32;  // wave M offset in block tile
  const int wn = (wave & 1) * 64;   // wave N offset in block tile

  // Per-thread fill geometry: 4 float4 slots (row, colgroup).
  int ldsoff[4];
  const float* px[4];
  const float* pw[4];
#pragma unroll
  for (int i = 0; i < 4; ++i) {
    const int s = t + i * 256;  // 1024 float4 slots
    const int r = s >> 3;       // row 0..127
    const int cg = (s & 7) * 4; // col group
    ldsoff[i] = r * 32 + cg;
    px[i] = X + (size_t)(m0 + r) * Dn + cg;
    pw[i] = W + (size_t)(n0 + r) * Dn + cg;
  }

  v8f acc[2][4];
#pragma unroll
  for (int mi = 0; mi < 2; ++mi)
#pragma unroll
    for (int ni = 0; ni < 4; ++ni) acc[mi][ni] = (v8f){};

  float4 rx[4], rw[4];
  // Prologue: load chunk 0, commit to buffer 0, issue loads for chunk 1.
#pragma unroll
  for (int i = 0; i < 4; ++i) {
    rx[i] = *(const float4*)(px[i]);
    rw[i] = *(const float4*)(pw[i]);
  }
#pragma unroll
  for (int i = 0; i < 4; ++i) {
    *(float4*)&sX[0][ldsoff[i]] = rx[i];
    *(float4*)&sW[0][ldsoff[i]] = rw[i];
  }
#pragma unroll
  for (int i = 0; i < 4; ++i) {
    rx[i] = *(const float4*)(px[i] + 32);
    rw[i] = *(const float4*)(pw[i] + 32);
  }
  __syncthreads();

  for (int it = 0; it < Dn / 32; ++it) {
    const float* cx = sX[it & 1];
    const float* cw = sW[it & 1];
#pragma unroll
    for (int kk = 0; kk < 8; ++kk) {
      const int kb = kk * 4 + lg * 2;  // 8-byte aligned
      v2f a0 = *(const v2f*)&cx[(wm + lm) * 32 + kb];
      v2f a1 = *(const v2f*)&cx[(wm + 16 + lm) * 32 + kb];
      v2f b0 = *(const v2f*)&cw[(wn + lm) * 32 + kb];
      v2f b1 = *(const v2f*)&cw[(wn + 16 + lm) * 32 + kb];
      v2f b2 = *(const v2f*)&cw[(wn + 32 + lm) * 32 + kb];
      v2f b3 = *(const v2f*)&cw[(wn + 48 + lm) * 32 + kb];
      acc[0][0] = WMMA_F32(a0, b0, acc[0][0]);
      acc[0][1] = WMMA_F32(a0, b1, acc[0][1]);
      acc[0][2] = WMMA_F32(a0, b2, acc[0][2]);
      acc[0][3] = WMMA_F32(a0, b3, acc[0][3]);
      acc[1][0] = WMMA_F32(a1, b0, acc[1][0]);
      acc[1][1] = WMMA_F32(a1, b1, acc[1][1]);
      acc[1][2] = WMMA_F32(a1, b2, acc[1][2]);
      acc[1][3] = WMMA_F32(a1, b3, acc[1][3]);
    }
    if (it < Dn / 32 - 1) {
      const int nb = (it + 1) & 1;
#pragma unroll
      for (int i = 0; i < 4; ++i) {
        *(float4*)&sX[nb][ldsoff[i]] = rx[i];
        *(float4*)&sW[nb][ldsoff[i]] = rw[i];
      }
      if (it < Dn / 32 - 2) {
        const int kc2 = (it + 2) * 32;
#pragma unroll
        for (int i = 0; i < 4; ++i) {
          rx[i] = *(const float4*)(px[i] + kc2);
          rw[i] = *(const float4*)(pw[i] + kc2);
        }
      }
      __syncthreads();
    }
  }

  // Epilogue: +bias, optional ReLU + residual. C layout: reg r, lane l ->
  // M = (l>=16)*8 + r, N = l%16.
#pragma unroll
  for (int ni = 0; ni < 4; ++ni) {
    const int n = n0 + wn + ni * 16 + lm;
    const float bv = bias[n];
#pragma unroll
    for (int mi = 0; mi < 2; ++mi) {
#pragma unroll
      for (int r = 0; r < 8; ++r) {
        const int m = m0 + wm + mi * 16 + lg * 8 + r;
        const size_t idx = (size_t)m * Dn + n;
        float val = acc[mi][ni][r] + bv;
        if (relu_res) val = fmaxf(val, 0.0f) + Res[idx];
        Out[idx] = val;
      }
    }
  }
}

// ---------------------------------------------------------------------------
// Attention: one block per (16-query tile, batch). Full 16x2048 score strip
// lives in dynamic LDS (~129KB; WGP has 320KB -> 2 blocks/WGP). 8 waves.
//   Phase 1: scores = Q*K^T / 32  -- 4 key tiles per iteration share each
//            Q A-fragment (4x Q-traffic reduction, 4 independent WMMA chains)
//   Phase 2: key-padding-masked softmax over the strip
//   Phase 3: O = P @ V / rowsum   -- 4 output chunks per iteration share each
//            P A-fragment
// ---------------------------------------------------------------------------
__global__ __launch_bounds__(256) void attn_flash(
    const float* __restrict__ Q, const float* __restrict__ K,
    const float* __restrict__ V, const int* __restrict__ lengths,
    float* __restrict__ O) {
  extern __shared__ float sm[];
  float* P = sm;               // [16][2048]
  float* red = sm + 16 * Sn;   // [256] partials + [16] rowmax + [16] 1/rowsum

  const int qt = blockIdx.x;
  const int b = blockIdx.y;
  const int len = lengths[b];
  const int t = threadIdx.x;
  const int wave = t >> 5;
  const int lane = t & 31;
  const int lm = lane & 15;
  const int lg = lane >> 4;

  const float* Qt = Q + ((size_t)b * Sn + qt * 16) * Dn;
  const float* Kb = K + (size_t)b * Sn * Dn;
  const float* Vb = V + (size_t)b * Sn * Dn;

  // ---- Phase 1: score strip (each wave: 16 key tiles = 4 groups of 4) ----
  const float* Qrow = Qt + (size_t)lm * Dn;
#pragma unroll 1
  for (int g = 0; g < 4; ++g) {
    const int ktb = wave * 16 + g * 4;  // first of 4 consecutive key tiles
    const float* Kr = Kb + (size_t)(ktb * 16 + lm) * Dn;
    v8f c0 = (v8f){}, c1 = (v8f){}, c2 = (v8f){}, c3 = (v8f){};
    for (int kk = 0; kk < Dn / 4; ++kk) {
      const int kb = kk * 4 + lg * 2;
      v2f a = *(const v2f*)(Qrow + kb);
      v2f b0 = *(const v2f*)(Kr + kb);
      v2f b1 = *(const v2f*)(Kr + (size_t)16 * Dn + kb);
      v2f b2 = *(const v2f*)(Kr + (size_t)32 * Dn + kb);
      v2f b3 = *(const v2f*)(Kr + (size_t)48 * Dn + kb);
      c0 = WMMA_F32(a, b0, c0);
      c1 = WMMA_F32(a, b1, c1);
      c2 = WMMA_F32(a, b2, c2);
      c3 = WMMA_F32(a, b3, c3);
    }
#pragma unroll
    for (int r = 0; r < 8; ++r) {
      float* Pr = P + (size_t)(lg * 8 + r) * Sn + ktb * 16 + lm;
      Pr[0]  = c0[r] * 0.03125f;  // 1/sqrt(1024)
      Pr[16] = c1[r] * 0.03125f;
      Pr[32] = c2[r] * 0.03125f;
      Pr[48] = c3[r] * 0.03125f;
    }
  }
  __syncthreads();

  // ---- Phase 2: masked softmax (16 rows x 16 segments of 128 cols) ----
  {
    const int row = t >> 4;
    const int seg = t & 15;
    float* Prow = P + (size_t)row * Sn;
    const int c0 = seg * 128;
    int cmax = c0 + 128;
    if (cmax > len) cmax = len;
    float mx = -3.402823e38f;
    for (int c = c0; c < cmax; ++c) mx = fmaxf(mx, Prow[c]);
    red[row * 16 + seg] = mx;
    __syncthreads();
    if (t < 16) {
      float mm = red[t * 16];
      for (int s2 = 1; s2 < 16; ++s2) mm = fmaxf(mm, red[t * 16 + s2]);
      red[256 + t] = mm;
    }
    __syncthreads();
    const float rmax = red[256 + row];
    float s = 0.0f;
    for (int c = c0; c < c0 + 128; ++c) {
      const float e = (c < len) ? __expf(Prow[c] - rmax) : 0.0f;
      Prow[c] = e;
      s += e;
    }
    red[row * 16 + seg] = s;
    __syncthreads();
    if (t < 16) {
      float ss = 0.0f;
      for (int s2 = 0; s2 < 16; ++s2) ss += red[t * 16 + s2];
      red[272 + t] = 1.0f / ss;
    }
    __syncthreads();
  }

  // ---- Phase 3: O = P @ V (each wave: 8 col chunks = 2 groups of 4) ----
  float* Ot = O + ((size_t)b * Sn + qt * 16) * Dn;
#pragma unroll 1
  for (int g = 0; g < 2; ++g) {
    const int n = (wave * 8 + g * 4) * 16 + lm;  // chunk j adds j*16
    v8f c0 = (v8f){}, c1 = (v8f){}, c2 = (v8f){}, c3 = (v8f){};
    for (int kt = 0; kt < Sn / 16; ++kt) {
#pragma unroll
      for (int ks = 0; ks < 4; ++ks) {
        const int kb = kt * 16 + ks * 4 + lg * 2;
        v2f a = *(const v2f*)(P + (size_t)lm * Sn + kb);  // ds_load_b64
        const float* Vr0 = Vb + (size_t)kb * Dn + n;
        const float* Vr1 = Vr0 + Dn;
        v2f b0; b0[0] = Vr0[0];  b0[1] = Vr1[0];
        v2f b1; b1[0] = Vr0[16]; b1[1] = Vr1[16];
        v2f b2; b2[0] = Vr0[32]; b2[1] = Vr1[32];
        v2f b3; b3[0] = Vr0[48]; b3[1] = Vr1[48];
        c0 = WMMA_F32(a, b0, c0);
        c1 = WMMA_F32(a, b1, c1);
        c2 = WMMA_F32(a, b2, c2);
        c3 = WMMA_F32(a, b3, c3);
      }
    }
#pragma unroll
    for (int r = 0; r < 8; ++r) {
      const int row = lg * 8 + r;
      const float rs = red[272 + row];
      float* Or = Ot + (size_t)row * Dn + n;
      Or[0]  = c0[r] * rs;
      Or[16] = c1[r] * rs;
      Or[32] = c2[r] * rs;
      Or[48] = c3[r] * rs;
    }
  }
}

// ---------------------------------------------------------------------------
// Fused (optional residual add) + LayerNorm over rows of length 1024.
// One 256-thread block per row; 4 elements per thread held in registers.
// ---------------------------------------------------------------------------
__global__ __launch_bounds__(256) void ln_kernel(
    const float* __restrict__ A, const float* __restrict__ R,
    const float* __restrict__ g, const float* __restrict__ be,
    float* __restrict__ out, int addres) {
  __shared__ float sred[8];
  const int row = blockIdx.x;
  const int t = threadIdx.x;
  const int lane = t & 31;
  const int wv = t >> 5;
  const size_t base = (size_t)row * Dn;

  float x[4];
  float s = 0.0f;
#pragma unroll
  for (int i = 0; i < 4; ++i) {
    const int c = t + i * 256;
    float v = A[base + c];
    if (addres) v += R[base + c];
    x[i] = v;
    s += v;
  }
#pragma unroll
  for (int o = 16; o > 0; o >>= 1) s += __shfl_xor(s, o, 32);
  if (lane == 0) sred[wv] = s;
  __syncthreads();
  float tot = 0.0f;
#pragma unroll
  for (int w = 0; w < 8; ++w) tot += sred[w];
  const float mean = tot * (1.0f / 1024.0f);
  __syncthreads();

  float vs = 0.0f;
#pragma unroll
  for (int i = 0; i < 4; ++i) {
    const float d = x[i] - mean;
    vs += d * d;
  }
#pragma unroll
  for (int o = 16; o > 0; o >>= 1) vs += __shfl_xor(vs, o, 32);
  if (lane == 0) sred[wv] = vs;
  __syncthreads();
  float vtot = 0.0f;
#pragma unroll
  for (int w = 0; w < 8; ++w) vtot += sred[w];
  const float rstd = rsqrtf(vtot * (1.0f / 1024.0f) + 1e-5f);

#pragma unroll
  for (int i = 0; i < 4; ++i) {
    const int c = t + i * 256;
    out[base + c] = (x[i] - mean) * rstd * g[c] + be[c];
  }
}

// ---------------------------------------------------------------------------
extern "C" void kernel_launch(void* const* d_in, const int* in_sizes, int n_in,
                              void* d_out, int out_size, void* d_ws,
                              size_t ws_size, hipStream_t stream) {
  (void)in_sizes; (void)n_in; (void)out_size; (void)ws_size;
  const float* seq = (const float*)d_in[0];
  const int* lengths = (const int*)d_in[1];
  const float* Wq = (const float*)d_in[2];
  const float* bq = (const float*)d_in[3];
  const float* Wk = (const float*)d_in[4];
  const float* bk = (const float*)d_in[5];
  const float* Wv = (const float*)d_in[6];
  const float* bv = (const float*)d_in[7];
  const float* Wo = (const float*)d_in[8];
  const float* bo = (const float*)d_in[9];
  const float* g1 = (const float*)d_in[10];
  const float* b1 = (const float*)d_in[11];
  const float* g2 = (const float*)d_in[12];
  const float* b2 = (const float*)d_in[13];
  float* out = (float*)d_out;
  float* ws = (float*)d_ws;

  const size_t NE = (size_t)Bn * Sn * Dn;  // 16M elements = 64MB
  float* Qb = ws;
  float* Kb = ws + NE;
  float* Vb = ws + 2 * NE;
  float* Ab = ws + 3 * NE;  // attention output
  float* Post = Qb;         // alias: Q dead after attention
  float* Ffn = Kb;          // alias: K dead after attention

  const dim3 ggrid(Bn * Sn / 128, Dn / 128);  // (128, 8)
  const dim3 gblk(256);

  gemm_wmma<<<ggrid, gblk, 0, stream>>>(seq, Wq, bq, nullptr, Qb, 0);
  gemm_wmma<<<ggrid, gblk, 0, stream>>>(seq, Wk, bk, nullptr, Kb, 0);
  gemm_wmma<<<ggrid, gblk, 0, stream>>>(seq, Wv, bv, nullptr, Vb, 0);

  const size_t smem = (size_t)(16 * Sn + 288) * sizeof(float);  // ~129KB
  hipFuncSetAttribute((const void*)attn_flash,
                      hipFuncAttributeMaxDynamicSharedMemorySize, (int)smem);
  attn_flash<<<dim3(Sn / 16, Bn), 256, smem, stream>>>(Qb, Kb, Vb, lengths, Ab);

  ln_kernel<<<Bn * Sn, 256, 0, stream>>>(seq, Ab, g1, b1, Post, 1);
  gemm_wmma<<<ggrid, gblk, 0, stream>>>(Post, Wo, bo, Post, Ffn, 1);
  ln_kernel<<<Bn * Sn, 256, 0, stream>>>(Ffn, nullptr, g2, b2, out, 0);
}